// Point2ImageProjection_29635274342533
// MI455X (gfx1250) — compile-verified
//
#include <hip/hip_runtime.h>

typedef __attribute__((ext_vector_type(2))) float v2f;
typedef __attribute__((ext_vector_type(8))) float v8f;

struct __attribute__((aligned(4))) F2 { float x, y; };   // align4: upper-half addr is only dword-aligned

#define BATCHES   8
#define MAXP      400000
#define NTOT      (BATCHES * MAXP)            // 3,200,000
#define OFF1      (NTOT * 2)                  // image_depths base   (6,400,000)
#define OFF2      (OFF1 + NTOT)               // batch_voxel base    (9,600,000)
#define OFF3      (OFF2 + NTOT * 3)           // point_mask base     (19,200,000)
#define ITERS     25                          // groups of 16 per wave
#define BLOCKS_X  125                         // 125 * 8 waves * 25 iters * 16 pts = 400,000
#define STRIDE    128                         // points advanced per block-iteration (8 waves * 16)

__global__ __launch_bounds__(256)
void p2i_proj_wmma(const int*   __restrict__ vc,      // (N,4) int32 [bidx, z, y, x]
                   const float* __restrict__ l2c,     // (B,4,4) f32
                   const float* __restrict__ k33,     // (B,3,3) f32
                   const int*   __restrict__ ishape,  // (B,2)  int32 [H, W]
                   float*       __restrict__ out)
{
  const int  lane = threadIdx.x & 31;
  const int  wave = threadIdx.x >> 5;
  const int  b    = blockIdx.y;
  const bool lo   = (lane < 16);

  // ---- Build per-lane A-matrix fragment of the combined 4x4 affine G ----
  // G rows: 0=proj_x, 1=proj_y, 2=proj_w, 3=depth; applied to [x,y,z,1].
  // A 16x4 layout: lane holds row (lane&15); lanes<16 carry K0,K1; lanes>=16 carry K2,K3.
  // Rows 0-3 = G, rows 8-11 = G again (duplicates results into lanes 16-31), rest zero.
  const int  row = lane & 15;
  const int  mr  = row & 3;
  const bool act = (row & 4) == 0;

  const float* L = l2c + b * 16;  // 4x4 row-major, rows 0..2 used
  const float* K = k33 + b * 9;

  float m0, m1, m2, m3;
  if (mr < 3) {
    const float k0 = K[mr * 3 + 0], k1 = K[mr * 3 + 1], k2 = K[mr * 3 + 2];
    m0 = k0 * L[0] + k1 * L[4] + k2 * L[8];
    m1 = k0 * L[1] + k1 * L[5] + k2 * L[9];
    m2 = k0 * L[2] + k1 * L[6] + k2 * L[10];
    m3 = k0 * L[3] + k1 * L[7] + k2 * L[11];
  } else {                        // depth row = lidar_to_cam row 2
    m0 = L[8]; m1 = L[9]; m2 = L[10]; m3 = L[11];
  }
  // fold VOXEL_SIZE and PC_MIN into the matrix
  const float g0 = m0 * 0.1f;
  const float g1 = m1 * 0.1f;
  const float g2 = m2 * 0.2f;
  const float g3 = m0 * -75.2f + m1 * -75.2f + m2 * -2.0f + m3;

  v2f A;
  A.x = act ? (lo ? g0 : g2) : 0.0f;
  A.y = act ? (lo ? g1 : g3) : 0.0f;

  const int H = ishape[b * 2 + 0];
  const int W = ishape[b * 2 + 1];

  const int4* __restrict__ recs = (const int4*)vc;

  // software-pipelined point stream; lanes n and n+16 share the same point index i
  int i = b * MAXP + (blockIdx.x * (ITERS * 8) + wave) * 16 + row;
  int4 rec = recs[i];

  for (int it = 0; it < ITERS; ++it) {
    // prefetch next iteration's record (uniform trip count -> EXEC stays full)
    const int  inext = (it + 1 < ITERS) ? (i + STRIDE) : i;
    const int4 recn  = recs[inext];

    // record = [bidx, z, y, x]; reference reorders to [bidx, x, y, z]
    const int xi = rec.w, yi = rec.z, zi = rec.y;

    // B 4x16 fragment: column = lane&15 (point), lanes<16 -> (x,y), lanes>=16 -> (z,1)
    v2f Bm;
    Bm.x = lo ? (float)xi : (float)zi;
    Bm.y = lo ? (float)yi : 1.0f;

    v8f C = {};
    v8f D = __builtin_amdgcn_wmma_f32_16x16x4_f32(
        /*neg_a=*/false, A, /*neg_b=*/false, Bm,
        /*c_mod=*/(short)0, C, /*reuse_a=*/false, /*reuse_b=*/false);

    // D[0..3]: lanes 0-15 see rows 0-3, lanes 16-31 see rows 8-11 (= duplicated rows 0-3)
    const float px  = D[0];
    const float py  = D[1];
    const float pw  = D[2];
    const float dep = D[3];

    const float den = (__builtin_fabsf(pw) > 1e-6f) ? pw : 1.0f;
    // single v_rcp_f32 instead of two IEEE divide sequences; results are truncated
    // to int right after, so the ~2ULP reciprocal error is immaterial.
    const float rin = __builtin_amdgcn_rcpf(den);
    const int   iu  = (int)(px * rin);             // trunc toward zero, like astype(int32)
    const int   iv  = (int)(py * rin);
    const int   gx  = (int)(0.5f * (float)iu);     // IMAGE_SCALE * float(int(u)) -> int
    const int   gy  = (int)(0.5f * (float)iv);

    const bool mask = (gx > 0) & (gx < W) & (gy > 0) & (gy < H) & (dep > 1.0f);

    // branch-free masked outputs (v_cndmask, keeps EXEC full for WMMA + stores)
    const float gxf   = mask ? (float)gx : 0.0f;
    const float gyf   = mask ? (float)gy : 0.0f;
    const float depm  = mask ? dep       : 0.0f;
    const float xf    = mask ? (float)xi : 0.0f;
    const float yf    = mask ? (float)yi : 0.0f;
    const float zf    = mask ? (float)zi : 0.0f;
    const float maskf = mask ? 1.0f      : 0.0f;

    // store-A: b64 pair — lower half: image_grid[2i..2i+1]; upper half: batch_voxel[3i..3i+1]
    F2 dA; dA.x = lo ? gxf : xf; dA.y = lo ? gyf : yf;
    const int offA = lo ? (2 * i) : (OFF2 + 3 * i);
    *(F2*)(out + offA) = dA;

    // store-B: b32 — lower half: depth[i]; upper half: voxel z
    const int offB = lo ? (OFF1 + i) : (OFF2 + 3 * i + 2);
    out[offB] = lo ? depm : zf;

    // store-C: b32 — all lanes write mask[i]; lanes n and n+16 write the identical
    // value to the identical address (merged by the store coalescer)
    out[OFF3 + i] = maskf;

    rec = recn;
    i   = inext;
  }
}

extern "C" void kernel_launch(void* const* d_in, const int* in_sizes, int n_in,
                              void* d_out, int out_size, void* d_ws, size_t ws_size,
                              hipStream_t stream) {
  const int*   vc     = (const int*)d_in[0];
  const float* l2c    = (const float*)d_in[1];
  const float* k33    = (const float*)d_in[2];
  const int*   ishape = (const int*)d_in[3];
  float*       out    = (float*)d_out;

  dim3 grid(BLOCKS_X, BATCHES, 1);
  p2i_proj_wmma<<<grid, 256, 0, stream>>>(vc, l2c, k33, ishape, out);
}